// CapsNet_40346922778906
// MI455X (gfx1250) — compile-verified
//
#include <hip/hip_runtime.h>
#include <math.h>

// ---- problem constants ----
#define B_    128
#define CIN   3
#define HIN   32
#define WIN   42
#define C1    32
#define H1y   9
#define W1x   13
#define C2    256
#define H2y   4
#define W2x   6
#define NCI   768   // input capsules
#define NCO   64    // output capsules
#define DIN   8
#define DOUT  16

#define WSTRIDE ((size_t)NCO * DOUT * DIN)   // route_w floats per i (8192)

typedef __attribute__((ext_vector_type(2))) float v2f;
typedef __attribute__((ext_vector_type(8))) float v8f;

// ---------------- conv1 + ReLU: (128,3,32,42) -> (128,32,9,13) ----------------
__global__ void k_conv1(const float* __restrict__ x, const float* __restrict__ w,
                        const float* __restrict__ bias, float* __restrict__ h1) {
  int idx = blockIdx.x * 256 + threadIdx.x;           // exact grid: 479232 threads
  int ox = idx % W1x; int t = idx / W1x;
  int oy = t % H1y;  t /= H1y;
  int oc = t % C1;   int b = t / C1;
  const float* xb = x + (size_t)b * CIN * HIN * WIN;
  const float* wc = w + (size_t)oc * CIN * 36;
  float acc = bias[oc];
  for (int ic = 0; ic < CIN; ++ic)
    for (int ky = 0; ky < 6; ++ky)
      for (int kx = 0; kx < 6; ++kx)
        acc = fmaf(xb[ic*HIN*WIN + (oy*3+ky)*WIN + (ox*3+kx)],
                   wc[ic*36 + ky*6 + kx], acc);
  h1[idx] = fmaxf(acc, 0.0f);
}

// -------- primary caps conv (s=2,k=3) + per-capsule squash -> u (128,768,8) --------
__global__ void k_pcaps(const float* __restrict__ h1, const float* __restrict__ w,
                        const float* __restrict__ bias, float* __restrict__ u) {
  __shared__ float sm[256];
  int idx = blockIdx.x * 256 + threadIdx.x;           // exact grid: 786432 threads
  int ox = idx % W2x; int t = idx / W2x;
  int oy = t % H2y;  t /= H2y;
  int oc = t % C2;   int b = t / C2;
  const float* hb = h1 + (size_t)b * C1 * H1y * W1x;
  const float* wc = w + (size_t)oc * C1 * 9;
  float acc = bias[oc];
  for (int ic = 0; ic < C1; ++ic) {
    const float* hp = hb + ic*H1y*W1x + (oy*2)*W1x + ox*2;
    const float* wp = wc + ic*9;
#pragma unroll
    for (int ky = 0; ky < 3; ++ky)
#pragma unroll
      for (int kx = 0; kx < 3; ++kx)
        acc = fmaf(hp[ky*W1x + kx], wp[ky*3 + kx], acc);
  }
  sm[threadIdx.x] = acc;
  __syncthreads();
  int base = threadIdx.x & ~7;                        // capsule = 8 consecutive elems
  float n2 = 0.0f;
#pragma unroll
  for (int j = 0; j < 8; ++j) { float v = sm[base + j]; n2 = fmaf(v, v, n2); }
  float scale = sqrtf(n2) / (1.0f + n2);              // norm/(1+norm^2)
  u[idx] = acc * scale;
}

// -------- routing softmax over o: c = softmax(b_logits, axis=o), wave per (b,i) row --------
__global__ void k_softmax(const float* __restrict__ bl, float* __restrict__ c) {
  int wid  = threadIdx.x >> 5;
  int lane = threadIdx.x & 31;
  int row  = blockIdx.x * 8 + wid;                    // 98304 rows total
  const float* p = bl + (size_t)row * NCO;
  float v0 = p[lane], v1 = p[lane + 32];
  float m = fmaxf(v0, v1);
#pragma unroll
  for (int s = 16; s > 0; s >>= 1) m = fmaxf(m, __shfl_xor(m, s, 32));
  float e0 = __expf(v0 - m), e1 = __expf(v1 - m);
  float ssum = e0 + e1;
#pragma unroll
  for (int s = 16; s > 0; s >>= 1) ssum += __shfl_xor(ssum, s, 32);
  float inv = 1.0f / ssum;
  float* q = c + (size_t)row * NCO;
  q[lane] = e0 * inv; q[lane + 32] = e1 * inv;
}

// -------- phase A: s_partial[chunk,b,o,k] = sum_{i in chunk} c[b,i,o] * xh[b,i,o,k]
// wave owns (16-sample b-tile, one o). c folds into A; sum_i rides in WMMA C chain.
// Pointer-bump induction: all three streams advance by loop-invariant strides so the
// unrolled copies lower to immediate-offset vmem off a single incremented base.
__global__ void k_route_s(const float* __restrict__ u, const float* __restrict__ W,
                          const float* __restrict__ c, float* __restrict__ sp) {
  int b0 = blockIdx.x * 16;                      // 8 b-tiles
  int o  = blockIdx.y * 8 + (threadIdx.x >> 5);  // 8 o-groups x 8 waves
  int i0 = blockIdx.z * 192;                     // 4 i-chunks
  int lane = threadIdx.x & 31;
  int m = lane & 15, h = lane >> 4;
  v8f acc = {};
  const float* up = u + (size_t)(b0 + m) * NCI * DIN + (size_t)i0 * DIN + 2*h;
  const float* cp = c + (size_t)(b0 + m) * NCI * NCO + (size_t)i0 * NCO + o;
  const float* wp = W + (((size_t)i0 * NCO + o) * DOUT + m) * DIN + 2*h;
#pragma unroll 4
  for (int ii = 0; ii < 192; ++ii) {
    if ((ii & 3) == 0)                            // speculative L0 prefetch, ~8 i ahead
      __builtin_prefetch(wp + 8 * WSTRIDE, 0, 0);
    float cv = *cp;
    v2f a0 = *(const v2f*)(up);                   // A: M=m, K = 2h + {0,1}
    v2f a1 = *(const v2f*)(up + 4);               //      K = 4 + 2h + {0,1}
    a0 *= cv; a1 *= cv;                           // fold c[b,i,o] into A rows
    v2f bb0 = *(const v2f*)(wp);                  // B: N=k=m, K = 2h + {0,1}
    v2f bb1 = *(const v2f*)(wp + 4);
    acc = __builtin_amdgcn_wmma_f32_16x16x4_f32(false, a0, false, bb0, (short)0, acc, false, false);
    acc = __builtin_amdgcn_wmma_f32_16x16x4_f32(false, a1, false, bb1, (short)0, acc, false, false);
    up += DIN; cp += NCO; wp += WSTRIDE;
  }
  float* spo = sp + (size_t)blockIdx.z * B_ * NCO * DOUT;
#pragma unroll
  for (int r = 0; r < 8; ++r) {                   // D row = r + 8h
    int brow = b0 + r + 8*h;
    spo[((size_t)brow * NCO + o) * DOUT + m] = acc[r];
  }
}

// -------- reduce i-chunk partials + squash -> v (and final output) --------
__global__ void k_vsquash(const float* __restrict__ sp, float* __restrict__ v,
                          float* __restrict__ outp, int writeOut) {
  __shared__ float sm[256];
  int g = blockIdx.x * 256 + threadIdx.x;   // 131072 elements
  float s = 0.0f;
#pragma unroll
  for (int ch = 0; ch < 4; ++ch) s += sp[(size_t)ch * B_ * NCO * DOUT + g];
  sm[threadIdx.x] = s;
  __syncthreads();
  int base = threadIdx.x & ~15;             // 16-wide capsule
  float n2 = 0.0f;
#pragma unroll
  for (int j = 0; j < 16; ++j) { float t = sm[base + j]; n2 = fmaf(t, t, n2); }
  float val = s * (sqrtf(n2) / (1.0f + n2));
  v[g] = val;
  if (writeOut) outp[g] = val;
}

// -------- phase B: b_logits[b,i,o] += sum_k xh[b,i,o,k] * v[b,o,k]
// recompute xh tile via WMMA, dot with v via half-wave shuffle reduction.
__global__ void k_route_b(const float* __restrict__ u, const float* __restrict__ W,
                          const float* __restrict__ v, float* __restrict__ bl) {
  int b0 = blockIdx.x * 16;
  int o  = blockIdx.y * 8 + (threadIdx.x >> 5);
  int i0 = blockIdx.z * 192;
  int lane = threadIdx.x & 31;
  int m = lane & 15, h = lane >> 4;
  float vl[8];
#pragma unroll
  for (int r = 0; r < 8; ++r)               // v tile: row (r+8h) x k=m
    vl[r] = v[((size_t)(b0 + r + 8*h) * NCO + o) * DOUT + m];
  const float* up = u + (size_t)(b0 + m) * NCI * DIN + (size_t)i0 * DIN + 2*h;
  const float* wp = W + (((size_t)i0 * NCO + o) * DOUT + m) * DIN + 2*h;
  float* bp = bl + (size_t)(b0 + m + 8*h) * NCI * NCO + (size_t)i0 * NCO + o;
#pragma unroll 4
  for (int ii = 0; ii < 192; ++ii) {
    if ((ii & 3) == 0)
      __builtin_prefetch(wp + 8 * WSTRIDE, 0, 0);
    v2f a0 = *(const v2f*)(up);
    v2f a1 = *(const v2f*)(up + 4);
    v2f bb0 = *(const v2f*)(wp);
    v2f bb1 = *(const v2f*)(wp + 4);
    v8f acc = {};
    acc = __builtin_amdgcn_wmma_f32_16x16x4_f32(false, a0, false, bb0, (short)0, acc, false, false);
    acc = __builtin_amdgcn_wmma_f32_16x16x4_f32(false, a1, false, bb1, (short)0, acc, false, false);
    float dl = 0.0f;
#pragma unroll
    for (int r = 0; r < 8; ++r) {           // sum over k = lanes of half-wave
      float t = acc[r] * vl[r];
      t += __shfl_xor(t, 1, 16);
      t += __shfl_xor(t, 2, 16);
      t += __shfl_xor(t, 4, 16);
      t += __shfl_xor(t, 8, 16);
      if (m == r) dl = t;                   // lane m<8 collects row (m+8h)
    }
    if (m < 8)                              // unique owner per (b,i,o): plain RMW
      *bp += dl;
    up += DIN; wp += WSTRIDE; bp += NCO;
  }
}

extern "C" void kernel_launch(void* const* d_in, const int* in_sizes, int n_in,
                              void* d_out, int out_size, void* d_ws, size_t ws_size,
                              hipStream_t stream) {
  (void)in_sizes; (void)n_in; (void)out_size; (void)ws_size;
  const float* x   = (const float*)d_in[0];
  const float* c1w = (const float*)d_in[1];
  const float* c1b = (const float*)d_in[2];
  const float* pcw = (const float*)d_in[3];
  const float* pcb = (const float*)d_in[4];
  const float* rw  = (const float*)d_in[5];
  float* out = (float*)d_out;

  // workspace carve-up (all f32), ~58 MB total
  char* ws = (char*)d_ws;
  size_t off = 0;
  auto carve = [&](size_t nfloats) {
    float* p = (float*)(ws + off);
    off = (off + nfloats * sizeof(float) + 255) & ~(size_t)255;
    return p;
  };
  float* h1 = carve((size_t)B_ * C1 * H1y * W1x);          //  1.9 MB
  float* u  = carve((size_t)B_ * NCI * DIN);               //  3.1 MB
  float* bl = carve((size_t)B_ * NCI * NCO);               // 25.2 MB
  float* c  = carve((size_t)B_ * NCI * NCO);               // 25.2 MB
  float* sp = carve((size_t)4 * B_ * NCO * DOUT);          //  2.1 MB
  float* v  = carve((size_t)B_ * NCO * DOUT);              //  0.5 MB

  k_conv1<<<(B_*C1*H1y*W1x)/256, 256, 0, stream>>>(x, c1w, c1b, h1);
  k_pcaps<<<(B_*C2*H2y*W2x)/256, 256, 0, stream>>>(h1, pcw, pcb, u);
  hipMemsetAsync(bl, 0, (size_t)B_ * NCI * NCO * sizeof(float), stream);

  for (int it = 0; it < 3; ++it) {
    k_softmax<<<(B_*NCI)/8, 256, 0, stream>>>(bl, c);
    k_route_s<<<dim3(8, 8, 4), 256, 0, stream>>>(u, rw, c, sp);
    k_vsquash<<<(B_*NCO*DOUT)/256, 256, 0, stream>>>(sp, v, out, it == 2 ? 1 : 0);
    if (it < 2)
      k_route_b<<<dim3(8, 8, 4), 256, 0, stream>>>(u, rw, v, bl);
  }
}